// CTCFormal_16896401342529
// MI455X (gfx1250) — compile-verified
//
#include <hip/hip_runtime.h>
#include <hip/hip_bf16.h>

// CTC forward loss, MI455X (gfx1250).
// Bandwidth-bound: 128 MB read / 23.3 TB/s ~= 5.5 us floor (L2-resident on
// graph replay: 128 MB < 192 MB global L2, so default RT caching, no NT hints).
// One wave32 per batch; rows staged global->LDS with gfx1250 async-to-LDS DMA
// (4-deep pipeline, constant-immediate s_wait_asynccnt, peeled drain), label
// gather from LDS, alpha recursion in 2 VGPRs/lane with one shuffle-up tap.

#define TT 64
#define NB 4096
#define CC 128
#define LL 31
#define WPB 8      // waves per block
#define DEPTH 4    // async pipeline depth (power of 2)

template <int W>
__device__ __forceinline__ void async_wait() {
  asm volatile("s_wait_asynccnt %0" :: "n"(W) : "memory");
}

__device__ __forceinline__ void async_load_b128(unsigned lds_addr, const float* gaddr) {
  // copies 16 bytes from global address gaddr to LDS address lds_addr; tracked by ASYNCcnt.
  asm volatile("global_load_async_to_lds_b128 %0, %1, off"
               :: "v"(lds_addr), "v"(gaddr)
               : "memory");
}

struct CtcLane {
  int   lane;
  int   lab;     // label owned by this lane (odd state 2k+1); 0 for lane>=31
  bool  skip;    // allow skip into odd state
  float a_lo;    // alpha[2k]   (blank state)
  float a_hi;    // alpha[2k+1] (label state)
};

// One recursion step consuming the LDS row `buf` (log-probs for time t).
__device__ __forceinline__ void ctc_step(CtcLane& c, const float* buf) {
  const float yb = __expf(buf[0]);            // blank prob (uniform DS read)
  float yl = __expf(buf[c.lab]);              // label prob (DS gather)
  yl = (c.lane < LL) ? yl : 0.0f;             // dummy state 63 stays dead
  float am1 = __shfl_up(c.a_hi, 1, 32);       // alpha[2k-1]
  am1 = (c.lane == 0) ? 0.0f : am1;
  const float nlo = (c.a_lo + am1) * yb;                        // even s: never skips
  const float nhi = (c.a_hi + c.a_lo + (c.skip ? am1 : 0.0f)) * yl; // odd s
  c.a_lo = nlo;
  c.a_hi = nhi;
}

__global__ __launch_bounds__(32 * WPB)
void ctc_alpha_kernel(const float* __restrict__ inp,        // (T, N, C) log-probs
                      const int* __restrict__ target,       // (N*L,)
                      const int* __restrict__ target_length,// (N,)
                      float* __restrict__ losses)           // (N,)
{
  __shared__ __align__(16) float smem[WPB][DEPTH][CC];

  CtcLane c;
  c.lane = threadIdx.x & 31;
  const int w = threadIdx.x >> 5;
  const int n = blockIdx.x * WPB + w;

  // Replicate reference's starts: 0 for n==0, else target_length[n-1].
  const int start = (n == 0) ? 0 : target_length[n - 1];
  c.lab = (c.lane < LL) ? target[start + c.lane] : 0;
  const int lab_prev = __shfl_up(c.lab, 1, 32);
  c.skip = (c.lane >= 1) && (c.lane < LL) && (c.lab != lab_prev);

  const float* row = inp + (size_t)n * CC;              // advances by N*C per t
  const size_t  tstride = (size_t)NB * CC;
  const unsigned lds0   = (unsigned)(uintptr_t)&smem[w][0][0];
  const unsigned lane16 = (unsigned)(c.lane << 4);

  // Prime the pipeline: rows 0..DEPTH-2 in flight.
  const float* pf = row + c.lane * 4;
#pragma unroll
  for (int t = 0; t < DEPTH - 1; ++t) {
    async_load_b128(lds0 + (unsigned)(t * CC * 4) + lane16, pf);
    pf += tstride;
  }

  // ---- t = 0: issue t=3, wait for row 0, initialize alpha. ----
  async_load_b128(lds0 + (unsigned)(3 * CC * 4) + lane16, pf);
  pf += tstride;
  async_wait<DEPTH - 1>();
  {
    const float* buf = &smem[w][0][0];
    const float yb = __expf(buf[0]);
    float yl = __expf(buf[c.lab]);
    yl = (c.lane < LL) ? yl : 0.0f;
    c.a_lo = (c.lane == 0) ? yb : 0.0f;     // alpha0[0] = y0[blank]
    c.a_hi = (c.lane == 0) ? yl : 0.0f;     // alpha0[1] = y0[label0]
  }

  // ---- main loop: t = 1 .. TT-DEPTH, constant wait <= DEPTH-1. ----
#pragma unroll 1
  for (int t = 1; t <= TT - DEPTH; ++t) {
    async_load_b128(lds0 + (unsigned)(((t + DEPTH - 1) & (DEPTH - 1)) * CC * 4) + lane16, pf);
    pf += tstride;
    async_wait<DEPTH - 1>();
    ctc_step(c, &smem[w][t & (DEPTH - 1)][0]);
  }

  // ---- peeled drain: t = TT-3, TT-2, TT-1 with tightening waits. ----
  async_wait<2>(); ctc_step(c, &smem[w][(TT - 3) & (DEPTH - 1)][0]);
  async_wait<1>(); ctc_step(c, &smem[w][(TT - 2) & (DEPTH - 1)][0]);
  async_wait<0>(); ctc_step(c, &smem[w][(TT - 1) & (DEPTH - 1)][0]);

  // loss = -log(alpha[62] + alpha[61]); alpha[62]=lane31.lo, alpha[61]=lane30.hi
  const float a61 = __shfl(c.a_hi, 30, 32);
  if (c.lane == 31) {
    losses[n] = -__logf(a61 + c.a_lo);
  }
}

__global__ __launch_bounds__(256)
void ctc_reduce_kernel(const float* __restrict__ losses, float* __restrict__ out) {
  __shared__ float red[256];
  float s = 0.0f;
  for (int i = threadIdx.x; i < NB; i += 256) s += losses[i];
  red[threadIdx.x] = s;
  __syncthreads();
  for (int off = 128; off > 0; off >>= 1) {
    if ((int)threadIdx.x < off) red[threadIdx.x] += red[threadIdx.x + off];
    __syncthreads();
  }
  if (threadIdx.x == 0) out[0] = red[0];
}

extern "C" void kernel_launch(void* const* d_in, const int* in_sizes, int n_in,
                              void* d_out, int out_size, void* d_ws, size_t ws_size,
                              hipStream_t stream) {
  const float* inp           = (const float*)d_in[0];  // (T,N,C) f32
  const int*   target        = (const int*)d_in[1];    // (N*L,) i32
  // d_in[2] = input_length: unused by the reference math (full-T scan)
  const int*   target_length = (const int*)d_in[3];    // (N,) i32
  float* losses = (float*)d_ws;                        // N floats of scratch
  float* out    = (float*)d_out;

  ctc_alpha_kernel<<<NB / WPB, 32 * WPB, 0, stream>>>(inp, target, target_length, losses);
  ctc_reduce_kernel<<<1, 256, 0, stream>>>(losses, out);
}